// Ours1_HS_Pers_67422396613012
// MI455X (gfx1250) — compile-verified
//
#include <hip/hip_runtime.h>
#include <math.h>

typedef __attribute__((ext_vector_type(2))) float v2f;
typedef __attribute__((ext_vector_type(8))) float v8f;

#define VSZ   512
#define BSZ   256
#define LSEQ  40
#define NPOS  (BSZ * (LSEQ - 1))        // 9984
#define NTILE (NPOS / 16)               // 624 tiles = 624 waves = 624 blocks

__global__ __launch_bounds__(32) void pers_nll_kernel(
    const int* __restrict__ seq, const float* __restrict__ sim,
    const float* __restrict__ pers, const float* __restrict__ wts,
    float* __restrict__ partial)
{
    __shared__ float maxArr[16];
    __shared__ float ztgArr[16];
    __shared__ float sumArr[16];

    const int lane    = threadIdx.x;     // 0..31, single wave per block
    const int tileIdx = blockIdx.x;

    const float w0 = wts[0];
    const float w1 = wts[1];

    // Per-lane position metadata (lanes 16-31 mirror lanes 0-15).
    const int m   = lane & 15;
    const int p   = tileIdx * 16 + m;              // always < NPOS (exact fit)
    const int b   = p / (LSEQ - 1);
    const int t   = p % (LSEQ - 1);
    const int cur = seq[b * LSEQ + t];
    const int nxt = seq[b * LSEQ + t + 1];
    const int prv = (t >= 1) ? seq[b * LSEQ + t - 1] : 0;
    const float w1e = (t >= 1) ? w1 : 0.0f;

    const float* srow = sim  + (size_t)cur * VSZ;
    const float* prow = pers + ((size_t)prv * VSZ + (size_t)cur) * VSZ;

    // ---- Phase 1: per-position row max (coalesced b128 stream) + target logit ----
    for (int q = 0; q < 16; ++q) {
        const int   cu  = __shfl(cur, q);
        const int   pv  = __shfl(prv, q);
        const int   nx  = __shfl(nxt, q);
        const float w1q = __shfl(w1e, q);
        const float* sr = sim  + (size_t)cu * VSZ;
        const float* pr = pers + ((size_t)pv * VSZ + (size_t)cu) * VSZ;
        float mx = -3.0e38f;
        for (int k = 0; k < 4; ++k) {
            const int off = (k * 32 + lane) * 4;   // 128 contiguous floats / step
            float4 s4 = *(const float4*)(sr + off);
            float4 p4 = *(const float4*)(pr + off);
            float z0 = w0 * s4.x + w1q * p4.x;
            float z1 = w0 * s4.y + w1q * p4.y;
            float z2 = w0 * s4.z + w1q * p4.z;
            float z3 = w0 * s4.w + w1q * p4.w;
            mx = fmaxf(mx, fmaxf(fmaxf(z0, z1), fmaxf(z2, z3)));
        }
        for (int d = 16; d >= 1; d >>= 1)          // wave32 reduction
            mx = fmaxf(mx, __shfl_xor(mx, d));
        if (lane == 0) {
            maxArr[q] = mx;
            ztgArr[q] = w0 * sr[nx] + w1q * pr[nx];
        }
    }
    __syncthreads();

    // ---- Phase 2: sumexp = E(16x512) x ones(512x16) via V_WMMA_F32_16X16X4_F32 ----
    // Each lane loads a contiguous float4 (b128) of its row; the two half-waves
    // cover 8 columns per step. Column->K mapping is free since sum order is
    // irrelevant: WMMA step A eats {.x,.y}, step B eats {.z,.w}.
    const float mym = maxArr[m];
    const float* sbase = srow + (lane >> 4) * 4;
    const float* pbase = prow + (lane >> 4) * 4;
    v2f bones; bones[0] = 1.0f; bones[1] = 1.0f;
    v8f c = {};

    float4 s4 = *(const float4*)(sbase);
    float4 p4 = *(const float4*)(pbase);
    for (int vb = 0; vb < VSZ - 8; vb += 8) {
        // prefetch next 8 columns while current exp/WMMA work executes
        float4 s4n = *(const float4*)(sbase + vb + 8);
        float4 p4n = *(const float4*)(pbase + vb + 8);
        v2f a0, a1;
        a0[0] = __expf(w0 * s4.x + w1e * p4.x - mym);
        a0[1] = __expf(w0 * s4.y + w1e * p4.y - mym);
        a1[0] = __expf(w0 * s4.z + w1e * p4.z - mym);
        a1[1] = __expf(w0 * s4.w + w1e * p4.w - mym);
        c = __builtin_amdgcn_wmma_f32_16x16x4_f32(false, a0, false, bones,
                                                  (short)0, c, false, false);
        c = __builtin_amdgcn_wmma_f32_16x16x4_f32(false, a1, false, bones,
                                                  (short)0, c, false, false);
        s4 = s4n; p4 = p4n;
    }
    {   // final 8 columns (already loaded)
        v2f a0, a1;
        a0[0] = __expf(w0 * s4.x + w1e * p4.x - mym);
        a0[1] = __expf(w0 * s4.y + w1e * p4.y - mym);
        a1[0] = __expf(w0 * s4.z + w1e * p4.z - mym);
        a1[1] = __expf(w0 * s4.w + w1e * p4.w - mym);
        c = __builtin_amdgcn_wmma_f32_16x16x4_f32(false, a0, false, bones,
                                                  (short)0, c, false, false);
        c = __builtin_amdgcn_wmma_f32_16x16x4_f32(false, a1, false, bones,
                                                  (short)0, c, false, false);
    }

    // C/D layout: VGPR r, lanes 0-15 -> M=r, lanes 16-31 -> M=8+r; all N equal.
    if (lane == 0)
        for (int r = 0; r < 8; ++r) sumArr[r] = c[r];
    if (lane == 16)
        for (int r = 0; r < 8; ++r) sumArr[8 + r] = c[r];
    __syncthreads();

    float contrib = 0.0f;
    if (lane < 16)
        contrib = logf(sumArr[m]) + maxArr[m] - ztgArr[m];
    for (int d = 16; d >= 1; d >>= 1)
        contrib += __shfl_xor(contrib, d);
    if (lane == 0) partial[tileIdx] = contrib;
}

__global__ __launch_bounds__(256) void reduce_kernel(
    const float* __restrict__ partial, float* __restrict__ out)
{
    __shared__ float buf[256];
    float acc = 0.0f;
    for (int i = threadIdx.x; i < NTILE; i += 256) acc += partial[i];  // fixed order
    buf[threadIdx.x] = acc;
    __syncthreads();
    for (int s = 128; s > 0; s >>= 1) {
        if ((int)threadIdx.x < s) buf[threadIdx.x] += buf[threadIdx.x + s];
        __syncthreads();
    }
    if (threadIdx.x == 0)
        out[0] = buf[0] + (float)BSZ * logf((float)VSZ);   // nll0 = B*log(V)
}

extern "C" void kernel_launch(void* const* d_in, const int* in_sizes, int n_in,
                              void* d_out, int out_size, void* d_ws, size_t ws_size,
                              hipStream_t stream) {
    const int*   seq  = (const int*)d_in[0];
    const float* sim  = (const float*)d_in[1];
    const float* pers = (const float*)d_in[2];
    const float* wts  = (const float*)d_in[3];
    float* partial = (float*)d_ws;                  // NTILE floats of scratch

    pers_nll_kernel<<<NTILE, 32, 0, stream>>>(seq, sim, pers, wts, partial);
    reduce_kernel<<<1, 256, 0, stream>>>(partial, (float*)d_out);
}